// Add_forward_85272280695302
// MI455X (gfx1250) — compile-verified
//
#include <hip/hip_runtime.h>

// Problem constants from the reference: B=1024, D=512, K=256, NC=1.
#define B_DIM 1024
#define D_DIM 512
#define K_DIM 256
#define T_DIM 1024   // augmented contraction dim = 2*D  ([x^2 | x])

typedef __attribute__((ext_vector_type(2))) float v2f;
typedef __attribute__((ext_vector_type(8))) float v8f;

// ---------------------------------------------------------------------------
// Kernel 1: per-class weight preparation.
//   WaT[t][k]  (t in [0,1024), k in [0,256)) stored transposed so the GEMM's
//   B-operand loads are contiguous-in-k (row-major in t).
//     WaT[d     ][k] = 0.5 * sig2[k][d]
//     WaT[512+d ][k] = -sig2[k][d] * means[k][d]
//   bias[k] = 0.5 * sum_d sig2[k][d] * means[k][d]^2   (LDS tree reduction)
// ---------------------------------------------------------------------------
__global__ __launch_bounds__(256) void gpdf_prep(const float* __restrict__ means,
                                                 const float* __restrict__ rho,
                                                 float* __restrict__ WaT,
                                                 float* __restrict__ bias) {
  const int k = blockIdx.x;     // one block per output class
  const int t = threadIdx.x;    // 256 threads, each handles D/256 = 2 features
  float partial = 0.0f;
#pragma unroll
  for (int i = 0; i < D_DIM / 256; ++i) {
    const int d = t + i * 256;
    const float m = means[k * D_DIM + d];
    const float r = rho[k * D_DIM + d];
    const float s = log1pf(expf(r));      // softplus, matches jnp.log1p(exp)
    const float s2 = s * s;
    WaT[d * K_DIM + k]           = 0.5f * s2;
    WaT[(D_DIM + d) * K_DIM + k] = -s2 * m;
    partial += 0.5f * s2 * m * m;
  }
  __shared__ float red[256];
  red[t] = partial;
  __syncthreads();
  for (int s = 128; s > 0; s >>= 1) {
    if (t < s) red[t] += red[t + s];
    __syncthreads();
  }
  if (t == 0) bias[k] = red[0];
}

// ---------------------------------------------------------------------------
// Kernel 2: out[b][k] = Xa[b][:] . WaT[:][k] + bias[k]  via V_WMMA_F32_16X16X4_F32.
//   Xa[b][t] = (t < 512) ? x[b][t]^2 : x[b][t-512]   (computed on the fly)
//   Block = 256 threads = 8 wave32; each wave owns one 16x16 output tile,
//   block covers 128 rows x 16 cols. Grid = (1024/128) x (256/16) = 8 x 16.
//
//   f32 WMMA operand layout (ISA 7.12.2):
//     A 16x4 : lanes 0-15 -> M=lane,  V0=K0, V1=K1 ; lanes 16-31 -> V0=K2, V1=K3
//     B 4x16 : lanes 0-15 -> N=lane,  V0=K0, V1=K1 ; lanes 16-31 -> V0=K2, V1=K3
//     C/D    : VGPR j: lanes 0-15 -> (M=j, N=lane), lanes 16-31 -> (M=j+8, N=lane-16)
// ---------------------------------------------------------------------------
__global__ __launch_bounds__(256) void gpdf_gemm_wmma(
    const float* __restrict__ x,     // [1024][512]
    const float* __restrict__ WaT,   // [1024][256]
    const float* __restrict__ bias,  // [256]
    float* __restrict__ out) {       // [1024][256]
  const int lane = threadIdx.x & 31;
  const int wave = threadIdx.x >> 5;     // 0..7
  const int half = lane >> 4;            // 0: lanes 0-15, 1: lanes 16-31
  const int mn   = lane & 15;            // M (for A) / N (for B) within tile
  const int row0 = blockIdx.x * 128 + wave * 16;
  const int col0 = blockIdx.y * 16;

  const float* __restrict__ xrow = x + (row0 + mn) * D_DIM;
  const int bcol = col0 + mn;

  v8f c = {};

  // Phase 1: t in [0,512): A = x^2, weights = 0.5*sig2
  for (int t0 = 0; t0 < D_DIM; t0 += 4) {
    const int ka = t0 + 2 * half;        // this lane-half holds K=ka, ka+1
    v2f a, b;
    const float a0 = xrow[ka];
    const float a1 = xrow[ka + 1];
    a.x = a0 * a0;
    a.y = a1 * a1;
    b.x = WaT[ka * K_DIM + bcol];
    b.y = WaT[(ka + 1) * K_DIM + bcol];
    c = __builtin_amdgcn_wmma_f32_16x16x4_f32(false, a, false, b,
                                              (short)0, c, false, false);
  }
  // Phase 2: t in [512,1024): A = x, weights = -sig2*means
  for (int t0 = 0; t0 < D_DIM; t0 += 4) {
    const int ka = t0 + 2 * half;
    v2f a, b;
    a.x = xrow[ka];
    a.y = xrow[ka + 1];
    b.x = WaT[(D_DIM + ka) * K_DIM + bcol];
    b.y = WaT[(D_DIM + ka + 1) * K_DIM + bcol];
    c = __builtin_amdgcn_wmma_f32_16x16x4_f32(false, a, false, b,
                                              (short)0, c, false, false);
  }

  const float bv = bias[bcol];
#pragma unroll
  for (int j = 0; j < 8; ++j) {
    out[(row0 + j + 8 * half) * K_DIM + bcol] = c[j] + bv;
  }
}

// ---------------------------------------------------------------------------
// Launch: d_in = {x, means, rho} (f32), d_out = (B, K) f32.
// Workspace: WaT = 1024*256 f32 (1 MiB) at ws+0, bias = 256 f32 after it.
// ---------------------------------------------------------------------------
extern "C" void kernel_launch(void* const* d_in, const int* in_sizes, int n_in,
                              void* d_out, int out_size, void* d_ws, size_t ws_size,
                              hipStream_t stream) {
  const float* x     = (const float*)d_in[0];
  const float* means = (const float*)d_in[1];
  const float* rho   = (const float*)d_in[2];
  float* out  = (float*)d_out;
  float* WaT  = (float*)d_ws;             // T_DIM * K_DIM floats = 1 MiB
  float* bias = WaT + T_DIM * K_DIM;      // 256 floats

  gpdf_prep<<<dim3(K_DIM), dim3(256), 0, stream>>>(means, rho, WaT, bias);
  gpdf_gemm_wmma<<<dim3(B_DIM / 128, K_DIM / 16), dim3(256), 0, stream>>>(
      x, WaT, bias, out);
}